// RnnDecoderWordsGuesser_54150947668571
// MI455X (gfx1250) — compile-verified
//
#include <hip/hip_runtime.h>
#include <math.h>

// Problem constants (match the reference)
#define HID   512
#define BATCH 1024
#define VOCAB 8192
#define TLEN  20
#define G3    1536   // 3 * HID

typedef __attribute__((ext_vector_type(2))) float v2f;
typedef __attribute__((ext_vector_type(8))) float v8f;

// fp32 WMMA: D(16x16) += A(16x4) * B(4x16), full fp32 (no precision loss).
__device__ __forceinline__ v8f wmma4(v2f a, v2f b, v8f c) {
  return __builtin_amdgcn_wmma_f32_16x16x4_f32(
      /*neg_a=*/false, a, /*neg_b=*/false, b,
      /*c_mod=*/(short)0, c, /*reuse_a=*/false, /*reuse_b=*/false);
}

__device__ __forceinline__ v2f ld2(const float* p) {
  return *(const v2f*)p;   // 8B-aligned by construction (even K offsets)
}

// A fragment: rows m0..m0+15 of row-major A (lda floats), K cols k0..k0+3.
// ISA layout: lanes 0-15 -> K = {k0, k0+1}; lanes 16-31 -> K = {k0+2, k0+3};
// row = lane & 15.  -> one contiguous float2 per lane.
__device__ __forceinline__ v2f afrag(const float* __restrict__ A, int lda,
                                     int m0, int k0, int lane) {
  int m = lane & 15, half = lane >> 4;
  return ld2(A + (size_t)(m0 + m) * lda + k0 + 2 * half);
}

// B = W^T where W is (N,K) row-major: B[k][n] = W[n][k].
// Symmetric layout: VGPR0 = rows K={k0,k0+2}, VGPR1 = K={k0+1,k0+3}; col = lane&15.
// -> contiguous float2 from row n of W.
__device__ __forceinline__ v2f bfrag(const float* __restrict__ W, int ldw,
                                     int n0, int k0, int lane) {
  int n = lane & 15, half = lane >> 4;
  return ld2(W + (size_t)(n0 + n) * ldw + k0 + 2 * half);
}

// Fast sigmoid: V_EXP_F32 + V_RCP_F32 (saturates correctly at +-inf).
__device__ __forceinline__ float fast_sigmoid(float x) {
  return __builtin_amdgcn_rcpf(1.0f + __expf(-x));
}

// Fast tanh: hardware V_TANH_F32 when the builtin exists, else exp+rcp
// identity tanh(x) = 1 - 2/(exp(2x)+1) (also saturates correctly).
__device__ __forceinline__ float fast_tanh(float x) {
#if __has_builtin(__builtin_amdgcn_tanhf)
  return __builtin_amdgcn_tanhf(x);
#else
  return 1.0f - 2.0f * __builtin_amdgcn_rcpf(__expf(2.0f * x) + 1.0f);
#endif
}

// ---------------------------------------------------------------------------
// One-time: gi0[g] = b_ih0[g] + sum_k relu(emb[BOS][k]) * w_ih0[g][k]
// (layer-0 input gates are constant across batch AND time)
// ---------------------------------------------------------------------------
__global__ void k_gi0(const float* __restrict__ emb,
                      const float* __restrict__ w_ih0,
                      const float* __restrict__ b_ih0,
                      float* __restrict__ gi0) {
  int g = blockIdx.x * blockDim.x + threadIdx.x;
  if (g >= G3) return;
  const float* w = w_ih0 + (size_t)g * HID;
  const float* e = emb + 1 * HID;          // BOS = 1
  float acc = b_ih0[g];
  for (int k = 0; k < HID; ++k)
    acc += fmaxf(e[k], 0.0f) * w[k];
  gi0[g] = acc;
}

// ---------------------------------------------------------------------------
// Split hidden_state_init (B, 2H) -> h0 (B,H), h1 (B,H)
// ---------------------------------------------------------------------------
__global__ void k_split(const float* __restrict__ hsi,
                        float* __restrict__ h0, float* __restrict__ h1) {
  int i = blockIdx.x * blockDim.x + threadIdx.x;
  if (i >= BATCH * HID) return;
  int b = i / HID, j = i % HID;
  h0[i] = hsi[(size_t)b * (2 * HID) + j];
  h1[i] = hsi[(size_t)b * (2 * HID) + HID + j];
}

// ---------------------------------------------------------------------------
// Layer-0 GRU step, fused GEMM(gh = h @ Whh0^T) + gate math.
// One wave -> 16x16 tile for each of the 3 gates (3 f32-WMMA accumulators).
// Grid: 2048 wave-tiles (64 M-tiles x 32 N-tiles) = 256 blocks x 8 waves.
// ---------------------------------------------------------------------------
__global__ void k_gru0(const float* __restrict__ h_in,
                       float* __restrict__ h_out,
                       const float* __restrict__ w_hh0,
                       const float* __restrict__ b_hh0,
                       const float* __restrict__ gi0) {
  int wave = (blockIdx.x * blockDim.x + threadIdx.x) >> 5;
  int lane = threadIdx.x & 31;
  int m0 = (wave >> 5) * 16;     // 64 M tiles
  int n0 = (wave & 31) * 16;     // 32 N tiles

  v8f cr = {}, cz = {}, cn = {};
#pragma unroll 2
  for (int k0 = 0; k0 < HID; k0 += 4) {
    v2f a  = afrag(h_in, HID, m0, k0, lane);
    v2f br = bfrag(w_hh0, HID, n0,             k0, lane);
    v2f bz = bfrag(w_hh0, HID, n0 + HID,       k0, lane);
    v2f bn = bfrag(w_hh0, HID, n0 + 2 * HID,   k0, lane);
    cr = wmma4(a, br, cr);
    cz = wmma4(a, bz, cz);
    cn = wmma4(a, bn, cn);
  }

  int nn = lane & 15, half = lane >> 4;
  int col = n0 + nn;
  float gir = gi0[col], giz = gi0[col + HID], gin = gi0[col + 2 * HID];
  float bhr = b_hh0[col], bhz = b_hh0[col + HID], bhn = b_hh0[col + 2 * HID];
#pragma unroll
  for (int r = 0; r < 8; ++r) {
    int row = m0 + r + 8 * half;
    float rg = fast_sigmoid(gir + cr[r] + bhr);
    float zg = fast_sigmoid(giz + cz[r] + bhz);
    float ng = fast_tanh(gin + rg * (cn[r] + bhn));
    float hp = h_in[(size_t)row * HID + col];
    h_out[(size_t)row * HID + col] = (1.0f - zg) * ng + zg * hp;
  }
}

// ---------------------------------------------------------------------------
// Layer-1 GRU step: two fused GEMMs (gi = x @ Wih1^T, gh = h @ Whh1^T),
// 6 f32-WMMA accumulators per wave, then gate math.
// ---------------------------------------------------------------------------
__global__ void k_gru1(const float* __restrict__ x,      // layer-0 new hidden
                       const float* __restrict__ h_in,
                       float* __restrict__ h_out,
                       const float* __restrict__ w_ih1,
                       const float* __restrict__ b_ih1,
                       const float* __restrict__ w_hh1,
                       const float* __restrict__ b_hh1) {
  int wave = (blockIdx.x * blockDim.x + threadIdx.x) >> 5;
  int lane = threadIdx.x & 31;
  int m0 = (wave >> 5) * 16;
  int n0 = (wave & 31) * 16;

  v8f ir = {}, iz = {}, in_ = {}, hr = {}, hz = {}, hn = {};
#pragma unroll 2
  for (int k0 = 0; k0 < HID; k0 += 4) {
    v2f ax = afrag(x,    HID, m0, k0, lane);
    v2f ah = afrag(h_in, HID, m0, k0, lane);
    ir  = wmma4(ax, bfrag(w_ih1, HID, n0,           k0, lane), ir);
    iz  = wmma4(ax, bfrag(w_ih1, HID, n0 + HID,     k0, lane), iz);
    in_ = wmma4(ax, bfrag(w_ih1, HID, n0 + 2 * HID, k0, lane), in_);
    hr  = wmma4(ah, bfrag(w_hh1, HID, n0,           k0, lane), hr);
    hz  = wmma4(ah, bfrag(w_hh1, HID, n0 + HID,     k0, lane), hz);
    hn  = wmma4(ah, bfrag(w_hh1, HID, n0 + 2 * HID, k0, lane), hn);
  }

  int nn = lane & 15, half = lane >> 4;
  int col = n0 + nn;
  float bir = b_ih1[col], biz = b_ih1[col + HID], bin = b_ih1[col + 2 * HID];
  float bhr = b_hh1[col], bhz = b_hh1[col + HID], bhn = b_hh1[col + 2 * HID];
#pragma unroll
  for (int r = 0; r < 8; ++r) {
    int row = m0 + r + 8 * half;
    float rg = fast_sigmoid(ir[r] + bir + hr[r] + bhr);
    float zg = fast_sigmoid(iz[r] + biz + hz[r] + bhz);
    float ng = fast_tanh(in_[r] + bin + rg * (hn[r] + bhn));
    float hp = h_in[(size_t)row * HID + col];
    h_out[(size_t)row * HID + col] = (1.0f - zg) * ng + zg * hp;
  }
}

// ---------------------------------------------------------------------------
// Vocab projection: logits = h1 @ w_out^T + b_out  (1024 x 8192, K=512).
// One wave -> 16 rows x 64 cols (4 accumulators, shared A fragment).
// Grid: 64 M-tiles x 128 N-groups = 8192 waves = 1024 blocks x 8 waves.
// Logits land in workspace (33.5 MB -> resident in 192 MB L2).
// ---------------------------------------------------------------------------
__global__ void k_logits(const float* __restrict__ x,
                         const float* __restrict__ w_out,
                         const float* __restrict__ b_out,
                         float* __restrict__ logits) {
  int wave = (blockIdx.x * blockDim.x + threadIdx.x) >> 5;
  int lane = threadIdx.x & 31;
  int m0 = (wave >> 7) * 16;      // 64 M tiles
  int n0 = (wave & 127) * 64;     // 128 N groups of 64

  v8f c0 = {}, c1 = {}, c2 = {}, c3 = {};
#pragma unroll 2
  for (int k0 = 0; k0 < HID; k0 += 4) {
    v2f a = afrag(x, HID, m0, k0, lane);
    c0 = wmma4(a, bfrag(w_out, HID, n0,      k0, lane), c0);
    c1 = wmma4(a, bfrag(w_out, HID, n0 + 16, k0, lane), c1);
    c2 = wmma4(a, bfrag(w_out, HID, n0 + 32, k0, lane), c2);
    c3 = wmma4(a, bfrag(w_out, HID, n0 + 48, k0, lane), c3);
  }

  int nn = lane & 15, half = lane >> 4;
  v8f acc[4] = {c0, c1, c2, c3};
#pragma unroll
  for (int j = 0; j < 4; ++j) {
    int col = n0 + 16 * j + nn;
    float bo = b_out[col];
#pragma unroll
    for (int r = 0; r < 8; ++r) {
      int row = m0 + r + 8 * half;
      logits[(size_t)row * VOCAB + col] = acc[j][r] + bo;
    }
  }
}

// ---------------------------------------------------------------------------
// log_softmax over V=8192 per batch row; reads L2-resident logits, writes
// the final (b, t, :) slice with NON-TEMPORAL stores (671 MB streamed once,
// never re-read -> keep it out of L2 so logits/weights stay resident).
// One block of 256 per row.
// ---------------------------------------------------------------------------
__global__ void k_lsm(const float* __restrict__ logits,
                      float* __restrict__ out, int t) {
  __shared__ float red[256];
  int b = blockIdx.x, tid = threadIdx.x;
  const float* row = logits + (size_t)b * VOCAB;

  float mx = -INFINITY;
  for (int v = tid; v < VOCAB; v += 256) mx = fmaxf(mx, row[v]);
  red[tid] = mx; __syncthreads();
  for (int s = 128; s > 0; s >>= 1) {
    if (tid < s) red[tid] = fmaxf(red[tid], red[tid + s]);
    __syncthreads();
  }
  mx = red[0]; __syncthreads();

  float sum = 0.0f;
  for (int v = tid; v < VOCAB; v += 256) sum += __expf(row[v] - mx);
  red[tid] = sum; __syncthreads();
  for (int s = 128; s > 0; s >>= 1) {
    if (tid < s) red[tid] += red[tid + s];
    __syncthreads();
  }
  float lse = mx + __logf(red[0]);

  float* o = out + ((size_t)b * TLEN + t) * VOCAB;
  for (int v = tid; v < VOCAB; v += 256)
    __builtin_nontemporal_store(row[v] - lse, &o[v]);
}

// ---------------------------------------------------------------------------
extern "C" void kernel_launch(void* const* d_in, const int* in_sizes, int n_in,
                              void* d_out, int out_size, void* d_ws, size_t ws_size,
                              hipStream_t stream) {
  const float* hsi   = (const float*)d_in[0];   // (B, 2H)
  const float* emb   = (const float*)d_in[1];   // (V, H)
  const float* w_ih  = (const float*)d_in[2];   // (2, 3H, H)
  const float* w_hh  = (const float*)d_in[3];   // (2, 3H, H)
  const float* b_ih  = (const float*)d_in[4];   // (2, 3H)
  const float* b_hh  = (const float*)d_in[5];   // (2, 3H)
  const float* w_out = (const float*)d_in[6];   // (V, H)
  const float* b_out = (const float*)d_in[7];   // (V,)
  float* out = (float*)d_out;                   // (B, T, V)

  // Workspace carve-up (all offsets 8B-aligned): ~42 MB total.
  float* ws     = (float*)d_ws;
  float* gi0    = ws;                       // 1536
  float* h0a    = gi0 + G3;
  float* h0b    = h0a + BATCH * HID;
  float* h1a    = h0b + BATCH * HID;
  float* h1b    = h1a + BATCH * HID;
  float* logits = h1b + BATCH * HID;        // B * V

  const float* w_hh0 = w_hh;
  const float* w_hh1 = w_hh + (size_t)G3 * HID;
  const float* w_ih1 = w_ih + (size_t)G3 * HID;
  const float* b_hh0 = b_hh;
  const float* b_hh1 = b_hh + G3;
  const float* b_ih1 = b_ih + G3;

  k_gi0<<<G3 / 256, 256, 0, stream>>>(emb, w_ih, b_ih, gi0);
  k_split<<<(BATCH * HID) / 256, 256, 0, stream>>>(hsi, h0a, h1a);

  float *h0c = h0a, *h0n = h0b, *h1c = h1a, *h1n = h1b;
  for (int t = 0; t < TLEN; ++t) {
    k_gru0<<<256, 256, 0, stream>>>(h0c, h0n, w_hh0, b_hh0, gi0);
    k_gru1<<<256, 256, 0, stream>>>(h0n, h1c, h1n, w_ih1, b_ih1, w_hh1, b_hh1);
    k_logits<<<1024, 256, 0, stream>>>(h1n, w_out, b_out, logits);
    k_lsm<<<BATCH, 256, 0, stream>>>(logits, out, t);
    float* tmp;
    tmp = h0c; h0c = h0n; h0n = tmp;
    tmp = h1c; h1c = h1n; h1n = tmp;
  }
}